// BioBERTEntityAware_26903675142736
// MI455X (gfx1250) — compile-verified
//
#include <hip/hip_runtime.h>
#include <hip/hip_bf16.h>

// Problem constants (from reference)
constexpr int BB = 64;    // batch
constexpr int SS = 512;   // sequence
constexpr int HH = 768;   // hidden
constexpr int EE = 128;   // entities
constexpr int TT = 4;     // tokens per entity

typedef __attribute__((ext_vector_type(2))) float v2f;
typedef __attribute__((ext_vector_type(8))) float v8f;

// ---------------------------------------------------------------------------
// Kernel 0: build selection matrix Msel (E x S), batch-invariant.
//   Msel[e][s] = 0.25 * (# of t with ent_tokens[e][t] == s)
// entity_embeddings[b] = Msel @ enhanced[b]  ==  gather + mean over T.
// Single block of 512 threads; adds of 0.25 are exact in fp32 => deterministic.
// ---------------------------------------------------------------------------
__global__ __launch_bounds__(512) void build_msel_kernel(
    const int* __restrict__ ent_tokens,  // (E*T)
    float* __restrict__ Msel)            // (E*S)
{
    const int tid = threadIdx.x;
    for (int i = tid; i < EE * SS; i += 512) Msel[i] = 0.0f;
    __syncthreads();
    if (tid < EE * TT) {
        const int e = tid >> 2;              // T == 4
        const int s = ent_tokens[tid];
        atomicAdd(&Msel[e * SS + s], 0.25f);
    }
}

// ---------------------------------------------------------------------------
// Kernel 1: enhanced = hidden + deterministic scatter-add of entity embeddings.
// One block per (b, s); 192 threads * float4 == 768 floats (one row).
// The E*T entry list is scanned into a 512-bit LDS mask so every thread
// accumulates matches in ascending entry order (bitwise deterministic), and
// multiplicity (duplicate (e,t) hits on the same s) is preserved per entry.
// No global atomics; single coalesced pass over the 100 MB tensor.
// ---------------------------------------------------------------------------
__global__ __launch_bounds__(192) void enhance_kernel(
    const float* __restrict__ hidden,        // (B,S,H)
    const int*   __restrict__ entity_types,  // (B,E)
    const float* __restrict__ entity_conf,   // (B,E)
    const int*   __restrict__ ent_tokens,    // (E,T)
    const float* __restrict__ type_table,    // (NUM_TYPES,H)
    const float* __restrict__ conf_w,        // (1,H)
    const float* __restrict__ conf_b,        // (H)
    float*       __restrict__ enhanced)      // (B,S,H) out
{
    __shared__ int      toks[EE * TT];
    __shared__ unsigned mask[(EE * TT) / 32];

    const int s   = blockIdx.x;
    const int b   = blockIdx.y;
    const int tid = threadIdx.x;

    for (int i = tid; i < EE * TT; i += 192) toks[i] = ent_tokens[i];
    if (tid < (EE * TT) / 32) mask[tid] = 0u;
    __syncthreads();

    for (int i = tid; i < EE * TT; i += 192)
        if (toks[i] == s) atomicOr(&mask[i >> 5], 1u << (i & 31));
    __syncthreads();

    const size_t row = ((size_t)b * SS + s) * (HH / 4);
    const float4* hin = (const float4*)hidden;
    float4*       out = (float4*)enhanced;
    const float4* tt4 = (const float4*)type_table;
    const float4* cw4 = (const float4*)conf_w;
    const float4* cb4 = (const float4*)conf_b;

    float4 v = hin[row + tid];

    const float4 cw = cw4[tid];
    const float4 cb = cb4[tid];

    for (int w = 0; w < (EE * TT) / 32; ++w) {
        unsigned m = mask[w];
        while (m) {
            const int bit = __ffs(m) - 1;
            m &= m - 1u;
            const int entry = w * 32 + bit;
            const int e     = entry >> 2;               // T == 4
            const int ty    = entity_types[b * EE + e];
            const float cf  = entity_conf[b * EE + e];
            const float4 te = tt4[(size_t)ty * (HH / 4) + tid];
            v.x += te.x + cf * cw.x + cb.x;
            v.y += te.y + cf * cw.y + cb.y;
            v.z += te.z + cf * cw.z + cb.z;
            v.w += te.w + cf * cw.w + cb.w;
        }
    }
    out[row + tid] = v;
}

// ---------------------------------------------------------------------------
// Kernel 2: entity_embeddings[b] = Msel(ExS) @ enhanced[b](SxH) via
// V_WMMA_F32_16X16X4_F32, one wave per 16x16 output tile, K=512 in steps of 4.
// VGPR layouts per CDNA5 ISA 7.12.2:
//   A (16x4 f32):  row M = lane%16 (both halves); VGPR v holds K = 2*(lane/16)+v
//   B (4x16 f32):  col N = lane%16; VGPR v holds K = 2*(lane/16)+v
//   C (16x16 f32): VGPR r -> M = r + 8*(lane/16), N = lane%16
// EXEC is all-ones (32-thread blocks, no divergence) as WMMA requires.
// ---------------------------------------------------------------------------
__global__ __launch_bounds__(32) void wmma_gather_mean_kernel(
    const float* __restrict__ enhanced,  // (B,S,H)
    const float* __restrict__ Msel,      // (E,S)
    float*       __restrict__ ent_emb)   // (B,E,H) out
{
    const int ntile = blockIdx.x;        // 0..H/16-1
    const int mtile = blockIdx.y;        // 0..E/16-1
    const int b     = blockIdx.z;        // 0..B-1
    const int lane  = threadIdx.x;       // 0..31
    const int g     = lane >> 4;         // lane-half
    const int r     = lane & 15;

    const int mbase = mtile * 16;
    const int nbase = ntile * 16;

    const float* __restrict__ Arow = Msel + (size_t)(mbase + r) * SS;        // A row for this lane
    const float* __restrict__ Bcol = enhanced + (size_t)b * SS * HH + nbase + r;

    v8f c = {};

    for (int kb = 0; kb < SS; kb += 4) {
        const int k0 = kb + 2 * g;
        // A: contiguous K pair from the selection-matrix row (8B load)
        const float2 ap = *(const float2*)(Arow + k0);
        v2f a; a.x = ap.x; a.y = ap.y;
        // B: two rows of enhanced[b], same column
        const float* bp = Bcol + (size_t)k0 * HH;
        v2f bv; bv.x = bp[0]; bv.y = bp[HH];

        c = __builtin_amdgcn_wmma_f32_16x16x4_f32(
                /*neg_a=*/false, a, /*neg_b=*/false, bv,
                /*c_mod=*/(short)0, c, /*reuse_a=*/false, /*reuse_b=*/false);
    }

    float* __restrict__ orow = ent_emb + ((size_t)b * EE + mbase) * HH + nbase + r;
#pragma unroll
    for (int i = 0; i < 8; ++i)
        orow[(size_t)(i + g * 8) * HH] = c[i];
}

// ---------------------------------------------------------------------------
extern "C" void kernel_launch(void* const* d_in, const int* in_sizes, int n_in,
                              void* d_out, int out_size, void* d_ws, size_t ws_size,
                              hipStream_t stream) {
    const float* hidden  = (const float*)d_in[0];  // (B,S,H) f32
    const int*   etypes  = (const int*)  d_in[1];  // (B,E)   i32
    const float* econf   = (const float*)d_in[2];  // (B,E)   f32
    const int*   etok    = (const int*)  d_in[3];  // (E,T)   i32
    const float* ttab    = (const float*)d_in[4];  // (5,H)   f32
    const float* cw      = (const float*)d_in[5];  // (1,H)   f32
    const float* cb      = (const float*)d_in[6];  // (H)     f32

    float* enhanced = (float*)d_out;                           // (B,S,H)
    float* ent_emb  = enhanced + (size_t)BB * SS * HH;         // (B,E,H)
    float* Msel     = (float*)d_ws;                            // E*S f32 = 256 KB

    // 0) batch-invariant selection matrix (deterministic: 0.25 sums are exact)
    build_msel_kernel<<<1, 512, 0, stream>>>(etok, Msel);

    // 1) fused copy + deterministic scatter-add
    dim3 g1(SS, BB);
    enhance_kernel<<<g1, 192, 0, stream>>>(hidden, etypes, econf, etok,
                                           ttab, cw, cb, enhanced);

    // 2) gather + mean as WMMA GEMM: ent_emb[b] = Msel @ enhanced[b]
    dim3 g2(HH / 16, EE / 16, BB);
    wmma_gather_mean_kernel<<<g2, 32, 0, stream>>>(enhanced, Msel, ent_emb);
}